// GeometricProductLayer_77111842832517
// MI455X (gfx1250) — compile-verified
//
#include <hip/hip_runtime.h>

// MI455X / gfx1250, wave32. Geometric product in Cl(3,0,1) as a sparse GEMM:
// per 16-element tile, OUT(16x16) = sum over 48 K-chunks of A_chunk(16x4) @ B_chunk(4x16)
// where A holds per-element products a_i*b_j and B holds the constant signed-permutation
// Cayley structure. Memory-bound (AI = 2 FLOP/B, ~402 MB stream, HBM floor ~17us);
// WMMA f32 16x16x4 does the contraction exactly in f32.

typedef __attribute__((ext_vector_type(2))) float v2f;
typedef __attribute__((ext_vector_type(8))) float v8f;

namespace {

constexpr int popc(int x) {
    int c = 0;
    while (x) { c += x & 1; x >>= 1; }
    return c;
}

// Cayley sign for blades A,B (bitmask over generators), metric (1,1,1,0):
// reorder sign from transposition count, zero when the degenerate generator
// (bit 3) is contracted. Matches the reference _build_cayley exactly.
constexpr float csign(int A, int B) {
    int s = 0;
    int sa = A >> 1;
    while (sa) { s += popc(sa & B); sa >>= 1; }
    float sgn = (s & 1) ? -1.0f : 1.0f;
    if (A & B & 8) sgn = 0.0f;
    return sgn;
}

// 48 chunks cover all 192 nonzero (i,j) pairs, 4 consecutive j per chunk:
//   c in [0,32):  i = c/4      (bit3 clear, all 16 j valid), j0 = (c%4)*4
//   c in [32,48): i = 8+(c-32)/2 (bit3 set, only j in 0..7), j0 = (c%2)*4
constexpr int chI(int c)  { return c < 32 ? (c >> 2) : 8 + ((c - 32) >> 1); }
constexpr int chJ0(int c) { return c < 32 ? ((c & 3) << 2) : ((c & 1) << 2); }

} // namespace

__global__ __launch_bounds__(256) void geo_prod_wmma(const float* __restrict__ a,
                                                     const float* __restrict__ b,
                                                     float* __restrict__ out,
                                                     int n_tiles) {
    const int lane = threadIdx.x & 31;
    const int hi   = lane >> 4;   // which K-half this lane supplies (K = 2*hi + reg)
    const int col  = lane & 15;   // A row / B column / D column

    // ---- Build the 48 constant B fragments once per wave (resident in VGPRs). ----
    // B layout (f32 4x16): VGPR0 -> K = 2*hi,  VGPR1 -> K = 2*hi + 1,  N = col.
    // B[K][n] = csign(i, j0+K) if n == i^(j0+K) else 0.
    float Bx[48], By[48];
#pragma unroll
    for (int c = 0; c < 48; ++c) {
        const int   i  = chI(c), j0 = chJ0(c);
        const int   t0a = i ^ (j0 + 0), t0b = i ^ (j0 + 2);
        const int   t1a = i ^ (j0 + 1), t1b = i ^ (j0 + 3);
        const float s0a = csign(i, j0 + 0), s0b = csign(i, j0 + 2);
        const float s1a = csign(i, j0 + 1), s1b = csign(i, j0 + 3);
        Bx[c] = hi ? ((col == t0b) ? s0b : 0.0f) : ((col == t0a) ? s0a : 0.0f);
        By[c] = hi ? ((col == t1b) ? s1b : 0.0f) : ((col == t1a) ? s1a : 0.0f);
    }

    const int wave   = blockIdx.x * (blockDim.x >> 5) + (threadIdx.x >> 5);
    const int nwaves = gridDim.x * (blockDim.x >> 5);

    for (int tile = wave; tile < n_tiles; tile += nwaves) {
        const size_t base = (size_t)tile * 256;  // dword offset of this 16x16 tile

        // Lane loads full 16-float rows for batch element (tile*16 + col).
        // Lanes 16..31 issue the same addresses as 0..15 in the same instruction;
        // the coalescer merges them, so DRAM traffic stays at the 128B/elem minimum.
        const float4* a4 = (const float4*)(a + base + (size_t)col * 16);
        const float4* b4 = (const float4*)(b + base + (size_t)col * 16);
        float as[16], bs[16];
        float4 t;
        t = a4[0]; as[0]=t.x;  as[1]=t.y;  as[2]=t.z;  as[3]=t.w;
        t = a4[1]; as[4]=t.x;  as[5]=t.y;  as[6]=t.z;  as[7]=t.w;
        t = a4[2]; as[8]=t.x;  as[9]=t.y;  as[10]=t.z; as[11]=t.w;
        t = a4[3]; as[12]=t.x; as[13]=t.y; as[14]=t.z; as[15]=t.w;
        t = b4[0]; bs[0]=t.x;  bs[1]=t.y;  bs[2]=t.z;  bs[3]=t.w;
        t = b4[1]; bs[4]=t.x;  bs[5]=t.y;  bs[6]=t.z;  bs[7]=t.w;
        t = b4[2]; bs[8]=t.x;  bs[9]=t.y;  bs[10]=t.z; bs[11]=t.w;
        t = b4[3]; bs[12]=t.x; bs[13]=t.y; bs[14]=t.z; bs[15]=t.w;

        // Pre-select the K-half of b once per tile (8 cndmasks instead of 96):
        // bsh[q] = bs[q + 2*hi] for the q values the chunks actually use.
        float bsh[16];
#pragma unroll
        for (int q = 0; q < 16; q += 4) {
            bsh[q + 0] = hi ? bs[q + 2] : bs[q + 0];
            bsh[q + 1] = hi ? bs[q + 3] : bs[q + 1];
        }

        v8f acc = {0.f, 0.f, 0.f, 0.f, 0.f, 0.f, 0.f, 0.f};
#pragma unroll
        for (int c = 0; c < 48; ++c) {
            const int i = chI(c), j0 = chJ0(c);
            // A layout (f32 16x4): VGPR0 -> K = 2*hi, VGPR1 -> K = 2*hi + 1, M = col.
            v2f A;
            A.x = as[i] * bsh[j0 + 0];   // W[m, 2*hi+0] = a[m,i] * b[m, j0+2*hi+0]
            A.y = as[i] * bsh[j0 + 1];   // W[m, 2*hi+1] = a[m,i] * b[m, j0+2*hi+1]
            v2f B;
            B.x = Bx[c];
            B.y = By[c];
            acc = __builtin_amdgcn_wmma_f32_16x16x4_f32(
                /*neg_a=*/false, A, /*neg_b=*/false, B,
                /*c_mod=*/(short)0, acc, /*reuse_a=*/false, /*reuse_b=*/false);
        }

        // D layout: VGPR v holds OUT[M = v + 8*hi, N = col].
        float* o = out + base + (size_t)(8 * hi) * 16 + col;
#pragma unroll
        for (int v = 0; v < 8; ++v) {
            o[(size_t)v * 16] = acc[v];
        }
    }
}

extern "C" void kernel_launch(void* const* d_in, const int* in_sizes, int n_in,
                              void* d_out, int out_size, void* d_ws, size_t ws_size,
                              hipStream_t stream) {
    const float* a = (const float*)d_in[0];
    const float* b = (const float*)d_in[1];
    // d_in[2] (cayley) is intentionally unused: the structure constants are
    // baked into the kernel at compile time (identical construction).
    float* out = (float*)d_out;

    const int n_tiles = in_sizes[0] / 256;  // 16 elements/tile * 16 blades each
    // 1024 blocks x 8 waves = 8192 waves -> 16 tiles/wave: amortizes the
    // one-time B-fragment build to ~2% while keeping enough waves in flight
    // to cover HBM latency on a pure streaming kernel.
    geo_prod_wmma<<<1024, 256, 0, stream>>>(a, b, out, n_tiles);
}